// ATSSAssigner_51376398795606
// MI455X (gfx1250) — compile-verified
//
#include <hip/hip_runtime.h>
#include <hip/hip_bf16.h>
#include <float.h>
#include <limits.h>
#include <stdint.h>

// ---------------------------------------------------------------------------
// ATSS assigner for MI455X (gfx1250, wave32).
// No matmul structure -> no WMMA. Strategy: exploit the V-shape of the
// per-level distance function so top-9 == contiguous 9-anchor window, and
// keep everything sparse (atomics into count/min-index arrays) instead of the
// reference's dense (B,M,N) tensors. Total traffic ~8 MB -> launch bound.
// CDNA5-specific paths used: global_load_async_to_lds_b64/b32 + s_wait_asynccnt
// for LDS staging, global_prefetch_b8 for the anchor windows.
// ---------------------------------------------------------------------------

#define ATSS_TOPK      9
#define ATSS_MAXLVL    8
#define ATSS_EPS       1e-9f
#define ATSS_ASYNC_LDS 1   // gfx1250 async memory->LDS staging (inline asm)

__device__ __forceinline__ float atss_ctr(const float* __restrict__ ab, int i) {
    // Matches reference: a_c = (a_s + a_e) * 0.5 computed from stored edges.
    return (ab[2 * i] + ab[2 * i + 1]) * 0.5f;
}

__device__ __forceinline__ float atss_iou(float gs, float ge, float as, float ae) {
    float inter = fminf(ge, ae) - fmaxf(gs, as);
    inter = fmaxf(inter, 0.0f);
    float uni = (ge - gs) + (ae - as) - inter;   // same association as reference
    return inter / (uni + ATSS_EPS);
}

// ---------------------------------------------------------------------------
// Kernel 1: init workspace (positive count per (b,n), min positive m per (b,n))
// ---------------------------------------------------------------------------
__global__ void atss_init_kernel(int* __restrict__ count, int* __restrict__ minm,
                                 int total) {
    int i = blockIdx.x * blockDim.x + threadIdx.x;
    if (i < total) {
        count[i] = 0;
        minm[i]  = INT_MAX;
    }
}

// ---------------------------------------------------------------------------
// Kernel 2: one thread per (b,m) GT box. Finds per-level 9-nearest windows,
// computes thr = mean + std(ddof=1) over the 4*9 candidate IoUs, and scatters
// positives via atomics.
// ---------------------------------------------------------------------------
__global__ void atss_candidates_kernel(const float* __restrict__ anchors,  // (N,2)
                                       const float* __restrict__ gtb,      // (B,M,2)
                                       const float* __restrict__ pad,      // (B,M)
                                       const int*   __restrict__ nal,      // (L)
                                       int B, int M, int N, int L,
                                       int* __restrict__ count,
                                       int* __restrict__ minm) {
    int tid = blockIdx.x * blockDim.x + threadIdx.x;
    if (tid >= B * M) return;
    const int b = tid / M;
    const int m = tid - b * M;

    const float gs = gtb[2 * tid];
    const float ge = gtb[2 * tid + 1];
    const bool  valid = pad[tid] > 0.0f;
    const float gc = (gs + ge) * 0.5f;

    // Per-level contiguous top-9 window starts (global anchor indices).
    int winLo[ATSS_MAXLVL];
    int nlev = (L < ATSS_MAXLVL) ? L : ATSS_MAXLVL;

    int off = 0;
    for (int l = 0; l < nlev; ++l) {
        const int nl = nal[l];
        // Analytic nearest-index guess from first two centers, then local
        // refine (float-exact argmin, ties -> smaller index).
        const float c0 = atss_ctr(anchors, off);
        const float c1 = atss_ctr(anchors, off + 1);
        const float st = c1 - c0;
        int guess = (int)floorf((gc - c0) / st + 0.5f);
        guess = min(max(guess, 0), nl - 1);
        __builtin_prefetch(&anchors[2 * (off + guess)], 0, 0);  // global_prefetch_b8

        int lo = max(guess - 2, 0), hi = min(guess + 2, nl - 1);
        int   best = lo;
        float bd   = fabsf(gc - atss_ctr(anchors, off + lo));
        for (int i = lo + 1; i <= hi; ++i) {
            float d = fabsf(gc - atss_ctr(anchors, off + i));
            if (d < bd) { bd = d; best = i; }
        }
        // Two-pointer expansion: picks the 9 smallest distances; on ties the
        // left (smaller index) wins, matching jax.lax.top_k stability.
        int Ls = best, Rs = best;
        for (int k = 1; k < ATSS_TOPK; ++k) {
            float dl = (Ls > 0)      ? fabsf(gc - atss_ctr(anchors, off + Ls - 1)) : FLT_MAX;
            float dr = (Rs < nl - 1) ? fabsf(gc - atss_ctr(anchors, off + Rs + 1)) : FLT_MAX;
            if (dl <= dr) --Ls; else ++Rs;
        }
        winLo[l] = off + Ls;   // window is [Ls, Ls+TOPK)
        off += nl;
    }

    const int K = nlev * ATSS_TOPK;

    // Pass 1: mean of gathered candidate IoUs (0 if gt invalid, like reference).
    float sum = 0.0f;
    for (int l = 0; l < nlev; ++l)
        for (int k = 0; k < ATSS_TOPK; ++k) {
            int a = winLo[l] + k;
            float v = valid ? atss_iou(gs, ge, anchors[2 * a], anchors[2 * a + 1]) : 0.0f;
            sum += v;
        }
    const float mean = sum / (float)K;

    // Pass 2: std with ddof=1.
    float ss = 0.0f;
    for (int l = 0; l < nlev; ++l)
        for (int k = 0; k < ATSS_TOPK; ++k) {
            int a = winLo[l] + k;
            float v = valid ? atss_iou(gs, ge, anchors[2 * a], anchors[2 * a + 1]) : 0.0f;
            float d = v - mean;
            ss += d * d;
        }
    const float thr = mean + sqrtf(ss / (float)(K - 1));

    // Pass 3: positives -> atomics. positive iff valid && iou>thr && center
    // strictly inside gt box (is_in_topk * is_in_gts * pad_gt_mask).
    for (int l = 0; l < nlev; ++l)
        for (int k = 0; k < ATSS_TOPK; ++k) {
            int a = winLo[l] + k;
            float as = anchors[2 * a], ae = anchors[2 * a + 1];
            float v = valid ? atss_iou(gs, ge, as, ae) : 0.0f;
            if (v > thr) {
                float ac = (as + ae) * 0.5f;
                if (valid && ac > gs && ac < ge) {
                    atomicAdd(&count[b * N + a], 1);
                    atomicMin(&minm[b * N + a], m);
                }
            }
        }
}

// ---------------------------------------------------------------------------
// Kernel 3: one thread per (b,n) anchor. GT boxes + labels staged in LDS via
// the gfx1250 async memory->LDS path (no VGPR round-trip).
//  count>1 : reference swaps in is_max_iou -> index = argmax_m iou (all m,
//            incl. padded gts; first-max tie-break), always foreground.
//  count==1: index = the unique positive m (minm).
//  count==0: index 0, background.
// ---------------------------------------------------------------------------
__global__ void atss_assign_kernel(const float* __restrict__ anchors,   // (N,2)
                                   const float* __restrict__ gtb,       // (B,M,2)
                                   const int*   __restrict__ glab,      // (B,M)
                                   const int*   __restrict__ bgp,       // scalar
                                   int B, int M, int N, int blocksPerB,
                                   const int* __restrict__ count,
                                   const int* __restrict__ minm,
                                   float* __restrict__ outL,            // (B,N)
                                   float* __restrict__ outBB,           // (B,N,2)
                                   float* __restrict__ outS) {          // (B,N,2)
    extern __shared__ unsigned char smem[];
    float* s_gt  = (float*)smem;            // 2*M floats (8B-aligned: offset 0)
    int*   s_lab = (int*)(s_gt + 2 * M);    // M ints

    const int b = blockIdx.x / blocksPerB;
    const int n = (blockIdx.x - b * blocksPerB) * blockDim.x + threadIdx.x;

#if ATSS_ASYNC_LDS
    // gfx1250: async copy from global memory to the LDS address in vdst, tracked by ASYNCcnt (ISA 10.7/15.18).
    // Generic shared pointer truncated to 32 bits == LDS byte address
    // (flat->LDS mapping uses addr[31:0]).
    for (int t = threadIdx.x; t < M; t += blockDim.x) {
        unsigned           lds_box = (unsigned)(uintptr_t)(const void*)&s_gt[2 * t];
        unsigned long long g_box   = (unsigned long long)(uintptr_t)&gtb[(b * M + t) * 2];
        asm volatile("global_load_async_to_lds_b64 %0, %1, off"
                     :: "v"(lds_box), "v"(g_box) : "memory");
        unsigned           lds_lab = (unsigned)(uintptr_t)(const void*)&s_lab[t];
        unsigned long long g_lab   = (unsigned long long)(uintptr_t)&glab[b * M + t];
        asm volatile("global_load_async_to_lds_b32 %0, %1, off"
                     :: "v"(lds_lab), "v"(g_lab) : "memory");
    }
    // Each issuing wave drains its own ASYNCcnt before the workgroup barrier.
    asm volatile("s_wait_asynccnt 0x0" ::: "memory");
#else
    for (int t = threadIdx.x; t < M; t += blockDim.x) {
        s_gt[2 * t]     = gtb[(b * M + t) * 2];
        s_gt[2 * t + 1] = gtb[(b * M + t) * 2 + 1];
        s_lab[t]        = glab[b * M + t];
    }
#endif
    __syncthreads();

    if (n >= N) return;
    const int idx = b * N + n;
    const int c   = count[idx];

    int  g   = 0;
    bool pos = false;
    if (c > 1) {
        const float as = anchors[2 * n], ae = anchors[2 * n + 1];
        float best = -FLT_MAX;
        int   bm   = 0;
        for (int mm = 0; mm < M; ++mm) {
            float v = atss_iou(s_gt[2 * mm], s_gt[2 * mm + 1], as, ae);
            if (v > best) { best = v; bm = mm; }   // first-max, like jnp.argmax
        }
        g = bm; pos = true;
    } else if (c == 1) {
        g = minm[idx]; pos = true;
    }

    const int lab = pos ? s_lab[g] : bgp[0];
    outL[idx] = (float)lab;

    float2 bb;
    bb.x = s_gt[2 * g];
    bb.y = s_gt[2 * g + 1];
    ((float2*)outBB)[idx] = bb;

    float2 sc;
    sc.x = (lab == 0) ? 1.0f : 0.0f;
    sc.y = (lab == 1) ? 1.0f : 0.0f;
    ((float2*)outS)[idx] = sc;
}

// ---------------------------------------------------------------------------
extern "C" void kernel_launch(void* const* d_in, const int* in_sizes, int n_in,
                              void* d_out, int out_size, void* d_ws, size_t ws_size,
                              hipStream_t stream) {
    const float* anchors = (const float*)d_in[0];   // (N,2) f32
    const int*   glab    = (const int*)  d_in[1];   // (B,M,1) i32
    const float* gtb     = (const float*)d_in[2];   // (B,M,2) f32
    const float* pad     = (const float*)d_in[3];   // (B,M,1) f32
    const int*   nal     = (const int*)  d_in[4];   // (L,) i32
    const int*   bgp     = (const int*)  d_in[5];   // scalar i32

    const int N  = in_sizes[0] / 2;
    const int BM = in_sizes[1];
    const int B  = out_size / (5 * N);   // outputs: B*N + 2*B*N + 2*B*N
    const int M  = BM / B;
    const int L  = in_sizes[4];

    int* count = (int*)d_ws;                 // B*N ints
    int* minm  = count + (size_t)B * N;      // B*N ints   (~2 MB total)

    const int total = B * N;
    atss_init_kernel<<<(total + 255) / 256, 256, 0, stream>>>(count, minm, total);

    const int pairs = B * M;
    atss_candidates_kernel<<<(pairs + 63) / 64, 64, 0, stream>>>(
        anchors, gtb, pad, nal, B, M, N, L, count, minm);

    const int blocksPerB = (N + 255) / 256;
    const size_t smem = (size_t)M * 3 * sizeof(float);   // 2*M floats + M ints
    float* outL  = (float*)d_out;
    float* outBB = outL + (size_t)B * N;
    float* outS  = outL + (size_t)3 * B * N;
    atss_assign_kernel<<<B * blocksPerB, 256, smem, stream>>>(
        anchors, gtb, glab, bgp, B, M, N, blocksPerB, count, minm, outL, outBB, outS);
}